// FinefyRelu_28664611733897
// MI455X (gfx1250) — compile-verified
//
#include <hip/hip_runtime.h>

typedef __attribute__((ext_vector_type(2))) float v2f;
typedef __attribute__((ext_vector_type(8))) float v8f;

#define KNB 9                    // neighbors (filter extent)
#define CIN 32
#define COUT 32
#define KTOT (KNB * CIN)         // 288 = GEMM reduction depth
#define LDS_STRIDE 290           // 288 + 2 pad: bank-conflict-free, keeps 8B alignment
#define WAVES_PER_BLOCK 8
#define BLOCK_THREADS (WAVES_PER_BLOCK * 32)

__global__ __launch_bounds__(BLOCK_THREADS)
void finefy_wmma_f32(const float* __restrict__ lv_coarse,
                     const int*   __restrict__ nbr,
                     const float* __restrict__ weight,   // [K, CIN, COUT] row-major = [288,32]
                     const float* __restrict__ bias,     // [COUT]
                     float*       __restrict__ out,      // [n_fine, COUT]
                     int n_fine, int n_tiles)
{
    // Weight transposed into LDS: wT[col * LDS_STRIDE + r] = W[r, col], r in [0,288), col in [0,32)
    __shared__ float wT[COUT * LDS_STRIDE];

    const int tid = threadIdx.x;
    for (int i = tid; i < KTOT * COUT; i += BLOCK_THREADS) {
        int r   = i / COUT;
        int col = i - r * COUT;
        wT[col * LDS_STRIDE + r] = weight[i];
    }
    __syncthreads();

    const int wave = tid >> 5;
    const int lane = tid & 31;
    const int half = lane >> 4;     // 0: lanes 0-15, 1: lanes 16-31
    const int mrow = lane & 15;     // M row within the 16-vertex tile
    const int koff = 2 * half;      // K-offset inside each 16x4 fragment (ISA A-layout)

    const int tile = blockIdx.x * WAVES_PER_BLOCK + wave;
    if (tile >= n_tiles) return;    // wave-uniform: EXEC stays all-1s for WMMA

    const int m0 = tile * 16;
    int m  = m0 + mrow;
    int mc = (m < n_fine) ? m : (n_fine - 1);   // clamp loads on tail tile

    // Per-lane neighbor indices for this vertex row (both wave halves load identical rows)
    int idxs[KNB];
    const int* ip = nbr + (long)mc * KNB;
#pragma unroll
    for (int k = 0; k < KNB; ++k) idxs[k] = ip[k];

    // Accumulators init = bias (C/D layout: lane holds column N = lane&15 / 16+(lane&15))
    v8f acc0, acc1;
    {
        float b0 = bias[mrow];
        float b1 = bias[16 + mrow];
#pragma unroll
        for (int r = 0; r < 8; ++r) { acc0[r] = b0; acc1[r] = b1; }
    }

#pragma unroll
    for (int k = 0; k < KNB; ++k) {
        const float* rp = lv_coarse + (long)idxs[k] * CIN;   // gathered coarse row (32 f32)
#pragma unroll
        for (int t = 0; t < CIN / 4; ++t) {
            // A fragment: lane (M=mrow) loads contiguous pair of K-cols -> global b64
            v2f a = *(const v2f*)(rp + 4 * t + koff);
            // B fragments: lane (N=col) loads contiguous pair of K-rows from transposed LDS
            const int r = k * CIN + 4 * t + koff;
            v2f b0 = *(const v2f*)&wT[mrow * LDS_STRIDE + r];
            v2f b1 = *(const v2f*)&wT[(16 + mrow) * LDS_STRIDE + r];
            acc0 = __builtin_amdgcn_wmma_f32_16x16x4_f32(
                false, a, false, b0, (short)0, acc0, false, false);
            acc1 = __builtin_amdgcn_wmma_f32_16x16x4_f32(
                false, a, false, b1, (short)0, acc1, false, false);
        }
    }

    // Fused ReLU + coalesced stores (lane -> output channel)
#pragma unroll
    for (int r = 0; r < 8; ++r) {
        int row = m0 + r + 8 * half;      // C/D layout: VGPR r holds M = r + 8*half
        if (row < n_fine) {
            float* o = out + (long)row * COUT;
            float v0 = acc0[r]; v0 = v0 > 0.0f ? v0 : 0.0f;
            float v1 = acc1[r]; v1 = v1 > 0.0f ? v1 : 0.0f;
            o[mrow]      = v0;
            o[16 + mrow] = v1;
        }
    }
}

extern "C" void kernel_launch(void* const* d_in, const int* in_sizes, int n_in,
                              void* d_out, int out_size, void* d_ws, size_t ws_size,
                              hipStream_t stream) {
    const float* lv_coarse = (const float*)d_in[0];
    const int*   nbr       = (const int*)  d_in[1];
    const float* weight    = (const float*)d_in[2];
    const float* bias      = (const float*)d_in[3];
    float*       out       = (float*)      d_out;

    const int n_fine  = out_size / COUT;               // 600000
    const int n_tiles = (n_fine + 15) / 16;            // 37500
    const int blocks  = (n_tiles + WAVES_PER_BLOCK - 1) / WAVES_PER_BLOCK;

    finefy_wmma_f32<<<blocks, BLOCK_THREADS, 0, stream>>>(
        lv_coarse, nbr, weight, bias, out, n_fine, n_tiles);
}